// GNNModel_10926396801112
// MI455X (gfx1250) — compile-verified
//
#include <hip/hip_runtime.h>
#include <hip/hip_bf16.h>
#include <stdint.h>

#define N_NODES 100000
#define N_EDGES 1600000
#define DFEAT   64

typedef __attribute__((ext_vector_type(2))) float v2f;
typedef __attribute__((ext_vector_type(8))) float v8f;

// ---------------------------------------------------------------------------
// deg[dst] += 1 for every edge (f32 so it feeds the mean division directly)
// ---------------------------------------------------------------------------
__global__ __launch_bounds__(256) void degree_kernel(
    const long long* __restrict__ dst, float* __restrict__ deg) {
  int e = blockIdx.x * blockDim.x + threadIdx.x;
  if (e >= N_EDGES) return;
  atomicAdd(&deg[(int)dst[e]], 1.0f);
}

// ---------------------------------------------------------------------------
// agg[dst, :] += feat[src, :]   (16 threads/edge, float4 gather, f32 atomics)
// The 25.6MB agg buffer lives in the 192MB L2, so the atomics stay on-chip.
// ---------------------------------------------------------------------------
__global__ __launch_bounds__(256) void scatter_kernel(
    const float* __restrict__ feat,
    const long long* __restrict__ src,
    const long long* __restrict__ dst,
    float* __restrict__ agg) {
  int tid = blockIdx.x * blockDim.x + threadIdx.x;
  int e = tid >> 4;
  if (e >= N_EDGES) return;
  int fb = (tid & 15) << 2;                  // feature sub-block of 4
  int s = (int)src[e];
  int d = (int)dst[e];
  const float4 v = *(const float4*)(feat + (long long)s * DFEAT + fb);
  float* a = agg + (long long)d * DFEAT + fb;
  atomicAdd(a + 0, v.x);
  atomicAdd(a + 1, v.y);
  atomicAdd(a + 2, v.z);
  atomicAdd(a + 3, v.w);
}

// ---------------------------------------------------------------------------
// out[16-node tile] = [relu]( (agg/deg) @ Wl^T + b + xin @ Wr^T )
// One wave32 per 16-node tile, f32 WMMA 16x16x4, K=64, 4 N-tiles of 16.
//   A  lane layout: M = lane%16, K = kb + 2*(lane/16) + {0,1}  -> float2 load
//   B  lane layout: N = lane%16, same K                        -> float2 load
//   C/D VGPR j    : M = j + 8*(lane/16), N = lane%16
// ---------------------------------------------------------------------------
template <bool RELU>
__global__ __launch_bounds__(256) void sage_node_kernel(
    const float* __restrict__ xin, const float* __restrict__ agg,
    const float* __restrict__ deg,
    const float* __restrict__ Wl, const float* __restrict__ bias,
    const float* __restrict__ Wr, float* __restrict__ out) {
  const int lane = threadIdx.x & 31;
  const int wave = threadIdx.x >> 5;
  const int ntiles = N_NODES / 16;                 // 6250 (N divisible by 16)
  const int tile = blockIdx.x * 8 + wave;
  if (tile >= ntiles) return;                      // wave-uniform: EXEC stays all-1s

  const int mlane = lane & 15;                     // A row / B,C,D col for this lane
  const int khalf = (lane >> 4) << 1;              // K sub-offset within 4-block: 0 or 2
  const int node  = tile * 16 + mlane;

  const float inv = 1.0f / fmaxf(deg[node], 1.0f); // mean aggregation scale

  v8f acc[4];
#pragma unroll
  for (int nb = 0; nb < 4; ++nb) {
    const float bv = bias[nb * 16 + mlane];        // col constant across all 8 rows
    v8f t = {bv, bv, bv, bv, bv, bv, bv, bv};
    acc[nb] = t;
  }

  const float* arow = agg + (long long)node * DFEAT;
  const float* xrow = xin + (long long)node * DFEAT;

#pragma unroll 4
  for (int kb = 0; kb < DFEAT; kb += 4) {
    const int k = kb + khalf;
    v2f a0 = *(const v2f*)(arow + k);
    a0 *= inv;                                     // fold mean-divide into A
    const v2f a1 = *(const v2f*)(xrow + k);
#pragma unroll
    for (int nb = 0; nb < 4; ++nb) {
      const int n = nb * 16 + mlane;
      const v2f bl = *(const v2f*)(Wl + n * DFEAT + k);   // B[k][n] = Wl[n][k]
      const v2f br = *(const v2f*)(Wr + n * DFEAT + k);
      acc[nb] = __builtin_amdgcn_wmma_f32_16x16x4_f32(
          false, a0, false, bl, (short)0, acc[nb], false, false);
      acc[nb] = __builtin_amdgcn_wmma_f32_16x16x4_f32(
          false, a1, false, br, (short)0, acc[nb], false, false);
    }
  }

#pragma unroll
  for (int nb = 0; nb < 4; ++nb) {
#pragma unroll
    for (int j = 0; j < 8; ++j) {
      const int m = j + ((lane >> 4) << 3);
      float v = acc[nb][j];
      if (RELU) v = fmaxf(v, 0.0f);
      out[(long long)(tile * 16 + m) * DFEAT + nb * 16 + mlane] = v;
    }
  }
}

// ---------------------------------------------------------------------------
// launch
// ---------------------------------------------------------------------------
extern "C" void kernel_launch(void* const* d_in, const int* in_sizes, int n_in,
                              void* d_out, int out_size, void* d_ws, size_t ws_size,
                              hipStream_t stream) {
  const float*     x   = (const float*)d_in[0];
  const long long* ei  = (const long long*)d_in[1];   // int64 per reference
  const float*     W1l = (const float*)d_in[2];
  const float*     b1  = (const float*)d_in[3];
  const float*     W1r = (const float*)d_in[4];
  const float*     W2l = (const float*)d_in[5];
  const float*     b2  = (const float*)d_in[6];
  const float*     W2r = (const float*)d_in[7];
  float* out = (float*)d_out;

  const long long* src = ei;             // edge_index[0]
  const long long* dst = ei + N_EDGES;   // edge_index[1]

  // workspace: [deg: N f32][agg: N*64 f32][h: N*64 f32]
  char*  ws        = (char*)d_ws;
  size_t agg_off   = ((size_t)N_NODES * 4 + 255) & ~(size_t)255;
  size_t agg_bytes = (size_t)N_NODES * DFEAT * 4;
  float* deg = (float*)(ws);
  float* agg = (float*)(ws + agg_off);
  float* h   = (float*)(ws + agg_off + agg_bytes);

  const int EB  = (N_EDGES + 255) / 256;            // degree grid
  const int SB  = (N_EDGES * 16 + 255) / 256;       // scatter grid
  const int NB  = (N_NODES / 16 + 7) / 8;           // node grid (8 waves/block)

  // deg + agg must be zero every call (graph replays don't re-poison)
  hipMemsetAsync(ws, 0, agg_off + agg_bytes, stream);

  degree_kernel<<<EB, 256, 0, stream>>>(dst, deg);

  // layer 1
  scatter_kernel<<<SB, 256, 0, stream>>>(x, src, dst, agg);
  sage_node_kernel<true><<<NB, 256, 0, stream>>>(x, agg, deg, W1l, b1, W1r, h);

  // layer 2
  hipMemsetAsync(ws + agg_off, 0, agg_bytes, stream);
  scatter_kernel<<<SB, 256, 0, stream>>>(h, src, dst, agg);
  sage_node_kernel<false><<<NB, 256, 0, stream>>>(h, agg, deg, W2l, b2, W2r, out);
}